// ModelNew_43173011259874
// MI455X (gfx1250) — compile-verified
//
#include <hip/hip_runtime.h>

typedef __attribute__((ext_vector_type(2))) float v2f;
typedef __attribute__((ext_vector_type(8))) float v8f;

#define LSTR 68   // LDS row stride in floats: 272B = 16B-aligned, conflict-free columns

#define Bn 2
#define Sn 4096
#define Hn 32
#define Pn 64
#define Nn 64
#define Ln 64
#define Cn 64     // Sn / Ln

// ---------------------------------------------------------------------------
// Pass 1: per-(b,c,h) chunk. Y_diag = (C B^T ∘ L) X ; raw states = X^T (decay⊙B)
// ---------------------------------------------------------------------------
__global__ __launch_bounds__(256) void ssd_chunk_kernel(
    const float* __restrict__ X, const float* __restrict__ A,
    const float* __restrict__ Bm, const float* __restrict__ Cm,
    float* __restrict__ Y, float* __restrict__ wsStates)
{
  __shared__ __align__(16) float sC[64 * LSTR];
  __shared__ __align__(16) float sB[64 * LSTR];
  __shared__ __align__(16) float sX[64 * LSTR];
  __shared__ float sCum[64];
  __shared__ float sDecay[64];
  __shared__ float sAv[64];

  const int gid = blockIdx.x;
  const int h = gid % Hn;
  const int c = (gid / Hn) % Cn;
  const int b = gid / (Hn * Cn);

  const int tid  = threadIdx.x;
  const int lane = tid & 31;
  const int wave = tid >> 5;
  const int half = lane >> 4;
  const int mr   = lane & 15;

  const size_t rowstride = (size_t)Hn * Pn;                  // between consecutive s
  const size_t base = ((size_t)b * Sn + (size_t)c * Ln) * rowstride + (size_t)h * Pn;

  // ---- stage 64x64 tiles of C, B, X into LDS (coalesced float4) ----
  {
    const int row = tid >> 2;
    const int q   = (tid & 3) * 16;
    const float* gx = X  + base + (size_t)row * rowstride + q;
    const float* gb = Bm + base + (size_t)row * rowstride + q;
    const float* gc = Cm + base + (size_t)row * rowstride + q;
    float* lx = sX + row * LSTR + q;
    float* lb = sB + row * LSTR + q;
    float* lc = sC + row * LSTR + q;
#pragma unroll
    for (int i = 0; i < 16; i += 4) {
      *(float4*)(lx + i) = *(const float4*)(gx + i);
      *(float4*)(lb + i) = *(const float4*)(gb + i);
      *(float4*)(lc + i) = *(const float4*)(gc + i);
    }
  }
  if (tid < 64)
    sAv[tid] = A[((size_t)b * Sn + (size_t)c * Ln + tid) * Hn + h];
  __syncthreads();
  if (tid == 0) {                      // inclusive cumsum over the chunk
    float s = 0.f;
    for (int l = 0; l < 64; ++l) { s += sAv[l]; sCum[l] = s; }
  }
  __syncthreads();
  if (tid < 64)
    sDecay[tid] = __expf(sCum[63] - sCum[tid]);
  __syncthreads();

  // ---- G = C * B^T (keep in regs), mask with L ----
  v8f g[2];
#pragma unroll
  for (int it = 0; it < 2; ++it) {
    const int t  = wave * 2 + it;
    const int ti = t >> 2, tj = t & 3;
    const float* arow = sC + (ti * 16 + mr) * LSTR;
    const float* brow = sB + (tj * 16 + mr) * LSTR;   // B^T: operand[k=n][s] = B[s][n]
    v8f acc = {};
#pragma unroll
    for (int kk = 0; kk < 64; kk += 4) {
      const int k = kk + 2 * half;
      v2f a;  a.x  = arow[k];  a.y  = arow[k + 1];
      v2f bv; bv.x = brow[k];  bv.y = brow[k + 1];
      acc = __builtin_amdgcn_wmma_f32_16x16x4_f32(false, a, false, bv, (short)0, acc,
                                                  false, false);
    }
    g[it] = acc;
  }
  __syncthreads();                      // all waves done reading sC
#pragma unroll
  for (int it = 0; it < 2; ++it) {
    const int t  = wave * 2 + it;
    const int ti = t >> 2, tj = t & 3;
#pragma unroll
    for (int r = 0; r < 8; ++r) {
      const int m = ti * 16 + r + 8 * half;   // l
      const int n = tj * 16 + mr;             // s
      float v = g[it][r];
      v = (m >= n) ? v * __expf(sCum[m] - sCum[n]) : 0.f;
      sC[m * LSTR + n] = v;                   // sC now holds masked M
    }
  }
  __syncthreads();

  // ---- Y_diag = M @ X ----
#pragma unroll
  for (int it = 0; it < 2; ++it) {
    const int t  = wave * 2 + it;
    const int ti = t >> 2, tj = t & 3;
    const float* arow = sC + (ti * 16 + mr) * LSTR;
    v8f acc = {};
#pragma unroll
    for (int kk = 0; kk < 64; kk += 4) {
      const int k = kk + 2 * half;
      v2f a;  a.x  = arow[k];                  a.y  = arow[k + 1];
      v2f bv; bv.x = sX[k * LSTR + tj * 16 + mr];
              bv.y = sX[(k + 1) * LSTR + tj * 16 + mr];
      acc = __builtin_amdgcn_wmma_f32_16x16x4_f32(false, a, false, bv, (short)0, acc,
                                                  false, false);
    }
#pragma unroll
    for (int r = 0; r < 8; ++r) {
      const int m = ti * 16 + r + 8 * half;   // l
      const int p = tj * 16 + mr;
      Y[base + (size_t)m * rowstride + p] = acc[r];
    }
  }

  // ---- raw states(p,n) = X^T @ (decay ⊙ B) ----
#pragma unroll
  for (int it = 0; it < 2; ++it) {
    const int t  = wave * 2 + it;
    const int ti = t >> 2, tj = t & 3;
    v8f acc = {};
#pragma unroll
    for (int kk = 0; kk < 64; kk += 4) {
      const int k = kk + 2 * half;
      v2f a;  a.x  = sX[k * LSTR + ti * 16 + mr];        // X^T: A[p][l] = X[l][p]
              a.y  = sX[(k + 1) * LSTR + ti * 16 + mr];
      v2f bv; bv.x = sB[k * LSTR + tj * 16 + mr] * sDecay[k];
              bv.y = sB[(k + 1) * LSTR + tj * 16 + mr] * sDecay[k + 1];
      acc = __builtin_amdgcn_wmma_f32_16x16x4_f32(false, a, false, bv, (short)0, acc,
                                                  false, false);
    }
    float* wsp = wsStates + (size_t)gid * 4096;          // [b][c][h][p*64+n]
#pragma unroll
    for (int r = 0; r < 8; ++r) {
      const int p = ti * 16 + r + 8 * half;
      const int n = tj * 16 + mr;
      wsp[p * 64 + n] = acc[r];
    }
  }
}

// ---------------------------------------------------------------------------
// Pass 2: per-(b,h) sequential inter-chunk scan, in place on workspace.
// S_in[0]=0 ; S_in[i+1] = exp(a_i)*S_in[i] + S_raw[i] ; ws[i] <- S_in[i]
// ---------------------------------------------------------------------------
__global__ __launch_bounds__(256) void ssd_scan_kernel(
    const float* __restrict__ A, float* __restrict__ ws)
{
  __shared__ float sE[64];
  const int gid = blockIdx.x;     // 0 .. Bn*Hn-1
  const int h = gid % Hn;
  const int b = gid / Hn;
  const int tid = threadIdx.x;

  if (tid < 64) {                 // exp of per-chunk A sums
    float s = 0.f;
    const float* ap = A + ((size_t)b * Sn + (size_t)tid * Ln) * Hn + h;
    for (int l = 0; l < 64; ++l) s += ap[(size_t)l * Hn];
    sE[tid] = __expf(s);
  }
  __syncthreads();

  for (int j = 0; j < 16; ++j) {
    const int elem = tid + j * 256;     // (p,n) flat index 0..4095
    float s_in = 0.f;
    for (int i = 0; i < 64; ++i) {
      const size_t idx = (((size_t)(b * Cn + i) * Hn + h) << 12) + elem;
      const float raw = ws[idx];
      ws[idx] = s_in;                   // state entering chunk i
      s_in = sE[i] * s_in + raw;
    }
  }
}

// ---------------------------------------------------------------------------
// Pass 3: Y += diag(exp(cum)) * C * S_in^T   per (b,c,h)
// ---------------------------------------------------------------------------
__global__ __launch_bounds__(256) void ssd_off_kernel(
    const float* __restrict__ A, const float* __restrict__ Cm,
    const float* __restrict__ wsStates, float* __restrict__ Y)
{
  __shared__ __align__(16) float sC[64 * LSTR];
  __shared__ __align__(16) float sS[64 * LSTR];
  __shared__ float sCum[64];
  __shared__ float sAv[64];

  const int gid = blockIdx.x;
  const int h = gid % Hn;
  const int c = (gid / Hn) % Cn;
  const int b = gid / (Hn * Cn);

  const int tid  = threadIdx.x;
  const int lane = tid & 31;
  const int wave = tid >> 5;
  const int half = lane >> 4;
  const int mr   = lane & 15;

  const size_t rowstride = (size_t)Hn * Pn;
  const size_t base = ((size_t)b * Sn + (size_t)c * Ln) * rowstride + (size_t)h * Pn;

  {
    const int row = tid >> 2;
    const int q   = (tid & 3) * 16;
    const float* gc = Cm + base + (size_t)row * rowstride + q;
    const float* gs = wsStates + (size_t)gid * 4096 + row * 64 + q;
    float* lc = sC + row * LSTR + q;
    float* ls = sS + row * LSTR + q;      // sS[p][n]
#pragma unroll
    for (int i = 0; i < 16; i += 4) {
      *(float4*)(lc + i) = *(const float4*)(gc + i);
      *(float4*)(ls + i) = *(const float4*)(gs + i);
    }
  }
  if (tid < 64)
    sAv[tid] = A[((size_t)b * Sn + (size_t)c * Ln + tid) * Hn + h];
  __syncthreads();
  if (tid == 0) {
    float s = 0.f;
    for (int l = 0; l < 64; ++l) { s += sAv[l]; sCum[l] = s; }
  }
  __syncthreads();

#pragma unroll
  for (int it = 0; it < 2; ++it) {
    const int t  = wave * 2 + it;
    const int ti = t >> 2, tj = t & 3;
    const float* arow = sC + (ti * 16 + mr) * LSTR;
    const float* srow = sS + (tj * 16 + mr) * LSTR;   // S^T: operand[k=n][p] = S[p][n]
    v8f acc = {};
#pragma unroll
    for (int kk = 0; kk < 64; kk += 4) {
      const int k = kk + 2 * half;
      v2f a;  a.x  = arow[k];  a.y  = arow[k + 1];
      v2f bv; bv.x = srow[k];  bv.y = srow[k + 1];
      acc = __builtin_amdgcn_wmma_f32_16x16x4_f32(false, a, false, bv, (short)0, acc,
                                                  false, false);
    }
#pragma unroll
    for (int r = 0; r < 8; ++r) {
      const int m = ti * 16 + r + 8 * half;   // l
      const int p = tj * 16 + mr;
      const size_t yi = base + (size_t)m * rowstride + p;
      Y[yi] += acc[r] * __expf(sCum[m]);
    }
  }
}

// ---------------------------------------------------------------------------
extern "C" void kernel_launch(void* const* d_in, const int* in_sizes, int n_in,
                              void* d_out, int out_size, void* d_ws, size_t ws_size,
                              hipStream_t stream)
{
  (void)in_sizes; (void)n_in; (void)out_size; (void)ws_size;
  const float* X  = (const float*)d_in[0];
  const float* A  = (const float*)d_in[1];
  const float* Bm = (const float*)d_in[2];
  const float* Cm = (const float*)d_in[3];
  float* Y        = (float*)d_out;
  float* wsStates = (float*)d_ws;     // Bn*Cn*Hn*4096 floats = 64 MB (fits in L2)

  dim3 blk(256);
  ssd_chunk_kernel<<<dim3(Bn * Cn * Hn), blk, 0, stream>>>(X, A, Bm, Cm, Y, wsStates);
  ssd_scan_kernel <<<dim3(Bn * Hn),      blk, 0, stream>>>(A, wsStates);
  ssd_off_kernel  <<<dim3(Bn * Cn * Hn), blk, 0, stream>>>(A, Cm, wsStates, Y);
}